// CoveragePlannerNet_3822520893845
// MI455X (gfx1250) — compile-verified
//
#include <hip/hip_runtime.h>

typedef __attribute__((ext_vector_type(2))) float v2f;
typedef __attribute__((ext_vector_type(8))) float v8f;

#define NAG 256           // agents per batch (N)
#define TAP_STRIDE 17     // 16 padded G rows + 1 pad float -> conflict-free LDS

// Fused CoveragePlannerNet: one block per batch element.
//  Stage A: per-agent compress MLP (FIN=32 -> 16 -> 8), build tap0 = z^T in LDS
//  Stage B: two graph shifts tap_{t+1} = tap_t @ S[b] via v_wmma_f32_16x16x4_f32
//  Stage C: per-agent graph filter (16x(3*8)) + relu + action head (5x16)
__global__ __launch_bounds__(256) void coverage_planner_kernel(
    const float* __restrict__ x,   const float* __restrict__ S,
    const float* __restrict__ W1,  const float* __restrict__ b1,
    const float* __restrict__ W2,  const float* __restrict__ b2,
    const float* __restrict__ Hgf, const float* __restrict__ bgf,
    const float* __restrict__ Wa,  const float* __restrict__ ba,
    float* __restrict__ out)
{
    // taps stored as T[t][n][g]: element (g-row, n-col) of the [16 x 256] tap matrix
    __shared__ float T[3][NAG * TAP_STRIDE];   // 3 * 256*17*4 = 52224 bytes

    const int bidx = blockIdx.x;
    const int tid  = threadIdx.x;

    // ---------------- Stage A: compress MLP, agent = tid ----------------
    {
        const float* xr = x + ((size_t)bidx * NAG + tid) * 32;
        float xi[32];
        #pragma unroll
        for (int i = 0; i < 32; ++i) xi[i] = xr[i];

        float h1[16];
        #pragma unroll
        for (int j = 0; j < 16; ++j) {
            float acc = b1[j];
            #pragma unroll
            for (int i = 0; i < 32; ++i) acc = fmaf(W1[j * 32 + i], xi[i], acc);
            h1[j] = fmaxf(acc, 0.0f);
        }
        #pragma unroll
        for (int g = 0; g < 8; ++g) {
            float acc = b2[g];
            #pragma unroll
            for (int i = 0; i < 16; ++i) acc = fmaf(W2[g * 16 + i], h1[i], acc);
            T[0][tid * TAP_STRIDE + g] = fmaxf(acc, 0.0f);
        }
        #pragma unroll
        for (int g = 8; g < 16; ++g)   // pad M rows 8..15 with zeros
            T[0][tid * TAP_STRIDE + g] = 0.0f;
    }
    __syncthreads();

    // ---------------- Stage B: K-tap graph shifts via WMMA ----------------
    const int wave = tid >> 5;
    const int lane = tid & 31;
    const int hi   = lane >> 4;      // half-wave select (K split for A/B frags)
    const int m    = lane & 15;
    const float* __restrict__ Sb = S + (size_t)bidx * NAG * NAG;
    const int n0 = wave * 32;        // this wave owns output columns [n0, n0+32)

    for (int t = 0; t < 2; ++t) {
        const float* __restrict__ A = T[t];
        v8f c0 = {}; v8f c1 = {};
        #pragma unroll 4
        for (int k0 = 0; k0 < NAG; k0 += 4) {
            const int r0 = k0 + 2 * hi;
            // A frag (16x4 f32): lanes 0-15 hold K=k0,k0+1; lanes 16-31 K=k0+2,k0+3
            v2f a;
            a.x = A[(r0    ) * TAP_STRIDE + m];
            a.y = A[(r0 + 1) * TAP_STRIDE + m];
            // B frags (4x16 f32) streamed straight from global S[b]
            v2f bA, bB;
            bA.x = Sb[(size_t)(r0    ) * NAG + n0 + m];
            bA.y = Sb[(size_t)(r0 + 1) * NAG + n0 + m];
            bB.x = Sb[(size_t)(r0    ) * NAG + n0 + 16 + m];
            bB.y = Sb[(size_t)(r0 + 1) * NAG + n0 + 16 + m];
            c0 = __builtin_amdgcn_wmma_f32_16x16x4_f32(false, a, false, bA,
                                                       (short)0, c0, false, false);
            c1 = __builtin_amdgcn_wmma_f32_16x16x4_f32(false, a, false, bB,
                                                       (short)0, c1, false, false);
        }
        // D layout: VGPR r -> row (r + 8*hi), col (n0[+16] + m); store as [n][g]
        float* __restrict__ D = T[t + 1];
        #pragma unroll
        for (int r = 0; r < 8; ++r) {
            D[(n0 + m)      * TAP_STRIDE + (r + 8 * hi)] = c0[r];
            D[(n0 + 16 + m) * TAP_STRIDE + (r + 8 * hi)] = c1[r];
        }
        __syncthreads();
    }

    // ---------------- Stage C: graph filter + action head, agent = tid ----------------
    {
        float y[16];
        #pragma unroll
        for (int f = 0; f < 16; ++f) {
            float acc = bgf[f];
            #pragma unroll
            for (int t = 0; t < 3; ++t) {
                #pragma unroll
                for (int g = 0; g < 8; ++g)
                    acc = fmaf(Hgf[(f * 3 + t) * 8 + g],
                               T[t][tid * TAP_STRIDE + g], acc);
            }
            y[f] = fmaxf(acc, 0.0f);
        }
        float* o = out + ((size_t)bidx * NAG + tid) * 5;
        #pragma unroll
        for (int a = 0; a < 5; ++a) {
            float acc = ba[a];
            #pragma unroll
            for (int f = 0; f < 16; ++f) acc = fmaf(Wa[a * 16 + f], y[f], acc);
            o[a] = acc;
        }
    }
}

extern "C" void kernel_launch(void* const* d_in, const int* in_sizes, int n_in,
                              void* d_out, int out_size, void* d_ws, size_t ws_size,
                              hipStream_t stream) {
    (void)in_sizes; (void)n_in; (void)out_size; (void)d_ws; (void)ws_size;
    const float* x   = (const float*)d_in[0];
    const float* S   = (const float*)d_in[1];
    const float* W1  = (const float*)d_in[2];
    const float* b1  = (const float*)d_in[3];
    const float* W2  = (const float*)d_in[4];
    const float* b2  = (const float*)d_in[5];
    const float* Hgf = (const float*)d_in[6];
    const float* bgf = (const float*)d_in[7];
    const float* Wa  = (const float*)d_in[8];
    const float* ba  = (const float*)d_in[9];

    coverage_planner_kernel<<<512, 256, 0, stream>>>(
        x, S, W1, b1, W2, b2, Hgf, bgf, Wa, ba, (float*)d_out);
}